// MinimizeEnergy_58067957842484
// MI455X (gfx1250) — compile-verified
//
#include <hip/hip_runtime.h>
#include <math.h>

// Problem constants (match reference setup_inputs()).
#define BATCH   4
#define NATOM   500000
#define NBOND_  500000
#define NANG_   700000
#define NTOR_   300000
#define NBEADS_ 125000
#define MSLOT   5
#define TB      256   // 8 wave32 waves per block

// ---------- small helpers ----------

struct V3 { float x, y, z; };
struct I3 { int x, y, z; };

static __device__ __forceinline__ V3 v3sub(V3 a, V3 b) { return {a.x-b.x, a.y-b.y, a.z-b.z}; }
static __device__ __forceinline__ V3 v3add(V3 a, V3 b) { return {a.x+b.x, a.y+b.y, a.z+b.z}; }
static __device__ __forceinline__ V3 v3scale(V3 a, float s) { return {a.x*s, a.y*s, a.z*s}; }
static __device__ __forceinline__ V3 v3neg(V3 a) { return {-a.x, -a.y, -a.z}; }
static __device__ __forceinline__ float v3dot(V3 a, V3 b) { return a.x*b.x + a.y*b.y + a.z*b.z; }
static __device__ __forceinline__ V3 v3cross(V3 a, V3 b) {
  return {a.y*b.z - a.z*b.y, a.z*b.x - a.x*b.z, a.x*b.y - a.y*b.x};
}
static __device__ __forceinline__ V3 loadp(const float* p) { return {p[0], p[1], p[2]}; }

static __device__ __forceinline__ float nan0(float x) {
  if (isnan(x)) return 0.0f;
  if (isinf(x)) return x > 0.0f ? 3.402823466e38f : -3.402823466e38f;
  return x;
}

// Native hardware float atomic add (global_atomic_add_f32 / _f64), relaxed, device scope.
static __device__ __forceinline__ void atomAddF(float* p, float v) {
  __hip_atomic_fetch_add(p, v, __ATOMIC_RELAXED, __HIP_MEMORY_SCOPE_AGENT);
}
static __device__ __forceinline__ void atomAddD(double* p, double v) {
  __hip_atomic_fetch_add(p, v, __ATOMIC_RELAXED, __HIP_MEMORY_SCOPE_AGENT);
}
static __device__ __forceinline__ void atomAdd3(float* p, V3 v) {
  atomAddF(p + 0, v.x);
  atomAddF(p + 1, v.y);
  atomAddF(p + 2, v.z);
}

// ---- gfx1250 async global->LDS staging (ASYNCcnt path), with safe fallback ----
typedef __attribute__((address_space(1))) int gint_t;   // global (AS1) int*
typedef __attribute__((address_space(3))) int lint_t;   // LDS (AS3) int*

static __device__ __forceinline__ void asyncCopyB32(const void* gsrc, void* ldst) {
#if __has_builtin(__builtin_amdgcn_global_load_async_to_lds_b32)
  // global_load_async_to_lds_b32: LDS gets data without a VGPR round-trip.
  __builtin_amdgcn_global_load_async_to_lds_b32(
      (gint_t*)(uintptr_t)gsrc,
      (lint_t*)(uint32_t)(uintptr_t)ldst,  // low 32 bits of generic addr == LDS offset
      0, 0);
#else
  *(uint32_t*)ldst = *(const uint32_t*)gsrc;
#endif
}
static __device__ __forceinline__ void asyncWaitZero() {
#if __has_builtin(__builtin_amdgcn_global_load_async_to_lds_b32)
# if __has_builtin(__builtin_amdgcn_s_wait_asynccnt)
  __builtin_amdgcn_s_wait_asynccnt(0);
# else
  asm volatile("s_wait_asynccnt 0x0" ::: "memory");
# endif
#endif
}

// wave32 shuffle reduce + LDS across waves -> one value at threadIdx.x==0.
static __device__ __forceinline__ float blockReduceSum(float v) {
  __shared__ float s[32];
  const int lane = threadIdx.x & 31;
  const int wid  = threadIdx.x >> 5;
  #pragma unroll
  for (int o = 16; o > 0; o >>= 1) v += __shfl_down(v, o, 32);
  if (lane == 0) s[wid] = v;
  __syncthreads();
  const int nw = (blockDim.x + 31) >> 5;
  v = (threadIdx.x < nw) ? s[threadIdx.x] : 0.0f;
  if (wid == 0) {
    #pragma unroll
    for (int o = 16; o > 0; o >>= 1) v += __shfl_down(v, o, 32);
  }
  return v;
}

// ---------- kernels ----------

__global__ void zero_kernel(float4* __restrict__ g4, double* __restrict__ esum) {
  const int i = blockIdx.x * blockDim.x + threadIdx.x;
  const int n4 = BATCH * NATOM * 3 / 4;  // 1,500,000 (exact)
  if (i < n4) g4[i] = make_float4(0.0f, 0.0f, 0.0f, 0.0f);
  if (i < 3) esum[i] = 0.0;
}

// E_b = 1000/(B*NB) * sum max((|p1-p0| - eq)^2 - tol^2, 0)
__global__ void bond_kernel(const float* __restrict__ pos, const int* __restrict__ idcs,
                            const float* __restrict__ eqv, const float* __restrict__ tolv,
                            float* __restrict__ g, double* __restrict__ esum) {
  int i = blockIdx.x * blockDim.x + threadIdx.x;
  float local = 0.0f;
  if (i < NBOND_) {
    int pf = i + 8192 < NBOND_ ? i + 8192 : NBOND_ - 1;
    __builtin_prefetch(eqv + pf, 0, 1);        // global_prefetch_b8 (streaming params)
    __builtin_prefetch(idcs + 2 * pf, 0, 1);
    const int2 ab = *(const int2*)(idcs + 2 * i);   // b64
    const int a0 = ab.x, a1 = ab.y;
    const float eq = eqv[i], tl = tolv[i];
    const float tl2 = tl * tl;
    const float kcoef = 2000.0f / ((float)BATCH * (float)NBOND_); // dE/dd = k*(d-eq)
    #pragma unroll
    for (int b = 0; b < BATCH; ++b) {
      const size_t o0 = ((size_t)b * NATOM + a0) * 3;
      const size_t o1 = ((size_t)b * NATOM + a1) * 3;
      const float dx = pos[o1+0] - pos[o0+0];
      const float dy = pos[o1+1] - pos[o0+1];
      const float dz = pos[o1+2] - pos[o0+2];
      const float d = sqrtf(dx*dx + dy*dy + dz*dz);
      const float diff = d - eq;
      const float act = diff * diff - tl2;
      if (act > 0.0f) {
        local += act;
        const float c = kcoef * diff / fmaxf(d, 1e-20f);
        atomAddF(g + o1 + 0,  c * dx); atomAddF(g + o1 + 1,  c * dy); atomAddF(g + o1 + 2,  c * dz);
        atomAddF(g + o0 + 0, -c * dx); atomAddF(g + o0 + 1, -c * dy); atomAddF(g + o0 + 2, -c * dz);
      }
    }
  }
  const float bs = blockReduceSum(local);
  if (threadIdx.x == 0 && bs != 0.0f) atomAddD(esum, (double)bs);
}

// E_a = 150/(B*NA) * sum max((acos(clip(cos)) - eq)^2 - tol^2, 0)
__global__ void angle_kernel(const float* __restrict__ pos, const int* __restrict__ idcs,
                             const float* __restrict__ eqv, const float* __restrict__ tolv,
                             float* __restrict__ g, double* __restrict__ esum) {
  int i = blockIdx.x * blockDim.x + threadIdx.x;
  float local = 0.0f;
  if (i < NANG_) {
    int pf = i + 8192 < NANG_ ? i + 8192 : NANG_ - 1;
    __builtin_prefetch(eqv + pf, 0, 1);
    __builtin_prefetch(idcs + 3 * pf, 0, 1);
    const I3 t3 = *(const I3*)(idcs + 3 * i);       // b96 (dword-aligned)
    const int a0 = t3.x, a1 = t3.y, a2 = t3.z;
    const float eq = eqv[i], tl = tolv[i];
    const float tl2 = tl * tl;
    const float kcoef = 300.0f / ((float)BATCH * (float)NANG_); // dE/da = k*diff
    const float lo = -1.0f + 1e-7f, hi = 1.0f - 1e-7f;
    #pragma unroll
    for (int b = 0; b < BATCH; ++b) {
      const float* base = pos + (size_t)b * NATOM * 3;
      const V3 p0 = loadp(base + (size_t)a0 * 3);
      const V3 p1 = loadp(base + (size_t)a1 * 3);
      const V3 p2 = loadp(base + (size_t)a2 * 3);
      const V3 b0 = v3sub(p0, p1);
      const V3 b1 = v3sub(p2, p1);
      const float n0 = sqrtf(v3dot(b0, b0));
      const float n1 = sqrtf(v3dot(b1, b1));
      const float denom = fmaxf(n0 * n1, 1e-9f);
      const float c = v3dot(b0, b1) / denom;
      const float ccl = fminf(fmaxf(c, lo), hi);
      const float a = acosf(ccl);
      const float diff = a - eq;
      const float act = diff * diff - tl2;
      if (act > 0.0f) {
        local += act;
        if (c > lo && c < hi) {  // clip passes gradient only in the interior
          const float dadc = -1.0f / sqrtf(fmaxf(1.0f - ccl * ccl, 1e-12f));
          const float dEdc = kcoef * diff * dadc;
          const float idn = 1.0f / denom;
          const float in02 = 1.0f / fmaxf(n0 * n0, 1e-18f);
          const float in12 = 1.0f / fmaxf(n1 * n1, 1e-18f);
          // dc/db0 = b1/denom - c*b0/n0^2 ; dc/db1 = b0/denom - c*b1/n1^2
          const V3 g0 = v3scale(v3sub(v3scale(b1, idn), v3scale(b0, c * in02)), dEdc);
          const V3 g2 = v3scale(v3sub(v3scale(b0, idn), v3scale(b1, c * in12)), dEdc);
          const V3 g1 = v3neg(v3add(g0, g2));
          atomAdd3(g + ((size_t)b * NATOM + a0) * 3, g0);
          atomAdd3(g + ((size_t)b * NATOM + a1) * 3, g1);
          atomAdd3(g + ((size_t)b * NATOM + a2) * 3, g2);
        }
      }
    }
  }
  const float bs = blockReduceSum(local);
  if (threadIdx.x == 0 && bs != 0.0f) atomAddD(esum, (double)bs);
}

// E_t = 100 * sum (2 - 2*cos(bounded)), bounded = sign(err)*max(|err|-tol,0),
// err = wrap(phi - eq), phi = atan2(m1.n2, n1.n2)  (praxeolitic dihedral).
__global__ void torsion_kernel(const float* __restrict__ pos, const int* __restrict__ idcs,
                               const float* __restrict__ eqv, const float* __restrict__ tolv,
                               float* __restrict__ g, double* __restrict__ esum) {
  const float PI = 3.14159265358979323846f;
  int i = blockIdx.x * blockDim.x + threadIdx.x;
  float local = 0.0f;
  if (i < NTOR_) {
    int pf = i + 8192 < NTOR_ ? i + 8192 : NTOR_ - 1;
    __builtin_prefetch(eqv + pf, 0, 1);
    __builtin_prefetch(idcs + 4 * pf, 0, 1);
    const int4 q = *(const int4*)(idcs + 4 * i);    // b128 (16B-aligned)
    const int a0 = q.x, a1 = q.y, a2 = q.z, a3 = q.w;
    const float eq = eqv[i], tl = tolv[i];
    #pragma unroll
    for (int b = 0; b < BATCH; ++b) {
      const float* base = pos + (size_t)b * NATOM * 3;
      const V3 p0 = loadp(base + (size_t)a0 * 3);
      const V3 p1 = loadp(base + (size_t)a1 * 3);
      const V3 p2 = loadp(base + (size_t)a2 * 3);
      const V3 p3 = loadp(base + (size_t)a3 * 3);
      const V3 b0 = v3sub(p1, p0);
      const V3 b1 = v3sub(p2, p1);
      const V3 b2 = v3sub(p3, p2);
      const V3 n1 = v3cross(b0, b1);
      const V3 n2 = v3cross(b1, b2);
      const float nb1 = sqrtf(v3dot(b1, b1));
      const float inb1 = 1.0f / fmaxf(nb1, 1e-9f);
      const V3 u = v3scale(b1, inb1);
      const V3 m1 = v3cross(n1, u);
      const float x = v3dot(n1, n2);
      const float y = v3dot(m1, n2);
      const float phi = atan2f(y, x);
      float err = phi - eq;
      err = fmodf(err + PI, 2.0f * PI);
      if (err < 0.0f) err += 2.0f * PI;   // python-style mod -> [0, 2pi)
      err -= PI;
      const float mag = fabsf(err) - tl;
      const float bounded = (mag > 0.0f) ? copysignf(mag, err) : 0.0f;
      local += 2.0f - 2.0f * cosf(bounded);   // 2 + cos(b-pi) + sin(b-pi/2)
      if (mag > 0.0f) {
        const float dEdphi = 200.0f * sinf(bounded);
        const float r2 = fmaxf(x * x + y * y, 1e-30f);
        const float f = dEdphi / r2;           // dphi = (x*dy - y*dx)/r2
        // dx/db*
        const V3 dxdb0 = v3cross(b1, n2);
        const V3 dxdb2 = v3cross(n1, b1);
        const V3 dxdb1 = v3add(v3cross(n2, b0), v3cross(b2, n1));
        // dy/db*  (w = u x n2; u-dependence gives (cn - u*y)/|b1| since u.(n2 x n1) == y)
        const V3 w = v3cross(u, n2);
        const V3 dydb0 = v3cross(b1, w);
        const V3 dydb2 = v3cross(m1, b1);
        const V3 cn = v3cross(n2, n1);
        const V3 term3 = v3scale(v3sub(cn, v3scale(u, y)), inb1);
        const V3 dydb1 = v3add(v3add(v3cross(w, b0), v3cross(b2, m1)), term3);
        const V3 Gb0 = v3scale(v3sub(v3scale(dydb0, x), v3scale(dxdb0, y)), f);
        const V3 Gb1 = v3scale(v3sub(v3scale(dydb1, x), v3scale(dxdb1, y)), f);
        const V3 Gb2 = v3scale(v3sub(v3scale(dydb2, x), v3scale(dxdb2, y)), f);
        const V3 gp0 = v3neg(Gb0);
        const V3 gp1 = v3sub(Gb0, Gb1);
        const V3 gp2 = v3sub(Gb1, Gb2);
        const V3 gp3 = Gb2;
        atomAdd3(g + ((size_t)b * NATOM + a0) * 3, gp0);
        atomAdd3(g + ((size_t)b * NATOM + a1) * 3, gp1);
        atomAdd3(g + ((size_t)b * NATOM + a2) * 3, gp2);
        atomAdd3(g + ((size_t)b * NATOM + a3) * 3, gp3);
      }
    }
  }
  const float bs = blockReduceSum(local);
  if (threadIdx.x == 0 && bs != 0.0f) atomAddD(esum, (double)bs);
}

// gradient = nan_to_num(-g); clip |F| to 0.1/dtau; pos = coords + gradient*dtau
__global__ void update_kernel(const float* __restrict__ coords, const float* __restrict__ g,
                              const float* __restrict__ dtau_p,
                              float* __restrict__ pos_out, float* __restrict__ grad_out) {
  int t = blockIdx.x * blockDim.x + threadIdx.x;
  if (t >= BATCH * NATOM) return;
  const float dtau = dtau_p[0];
  const size_t o = (size_t)t * 3;
  float gx = nan0(-g[o+0]);
  float gy = nan0(-g[o+1]);
  float gz = nan0(-g[o+2]);
  const float fn = sqrtf(gx*gx + gy*gy + gz*gz);
  const float limit = 0.1f / dtau;
  const float sc = (fn > limit) ? (limit / fmaxf(fn, 1e-12f)) : 1.0f;
  gx *= sc; gy *= sc; gz *= sc;
  grad_out[o+0] = gx; grad_out[o+1] = gy; grad_out[o+2] = gz;
  pos_out[o+0] = coords[o+0] + gx * dtau;
  pos_out[o+1] = coords[o+1] + gy * dtau;
  pos_out[o+2] = coords[o+2] + gz * dtau;
}

// rec[b,n,:] = bead[b,n,:] - sum_m w[n,m]*nan_to_num(pos[b,idx[n,m],:])  (gather BEFORE scatter)
// Bead tables staged block-wide into LDS via async global->LDS copies.
__global__ void bead_rec_kernel(const float* __restrict__ pos, const int* __restrict__ bidc,
                                const float* __restrict__ wgt, const float* __restrict__ beadc,
                                float* __restrict__ rec) {
  __shared__ int   s_idx[TB * MSLOT];
  __shared__ float s_w  [TB * MSLOT];
  const int tid  = threadIdx.x;
  const int blk0 = blockIdx.x * TB;                          // first bead of this block
  const int nb   = (NBEADS_ - blk0 < TB) ? (NBEADS_ - blk0) : TB;
  const int ne   = nb * MSLOT;
  for (int k = tid; k < ne; k += TB) {
    asyncCopyB32(bidc + (size_t)blk0 * MSLOT + k, &s_idx[k]);
    asyncCopyB32(wgt  + (size_t)blk0 * MSLOT + k, &s_w[k]);
  }
  asyncWaitZero();
  __syncthreads();
  const int n = blk0 + tid;
  if (n >= NBEADS_) return;
  #pragma unroll
  for (int b = 0; b < BATCH; ++b) {
    float sx = 0.0f, sy = 0.0f, sz = 0.0f;
    #pragma unroll
    for (int m = 0; m < MSLOT; ++m) {
      const int   idx = s_idx[tid * MSLOT + m];
      const float wm  = s_w  [tid * MSLOT + m];
      if (idx >= 0) {    // padded slots carry zero weight
        const float* p = pos + ((size_t)b * NATOM + idx) * 3;
        sx += wm * nan0(p[0]);
        sy += wm * nan0(p[1]);
        sz += wm * nan0(p[2]);
      }
    }
    const size_t ro = ((size_t)b * NBEADS_ + n) * 3;
    rec[ro+0] = beadc[ro+0] - sx;
    rec[ro+1] = beadc[ro+1] - sy;
    rec[ro+2] = beadc[ro+2] - sz;
  }
}

// pos[b, idx[n,m], :] += rec[b,n,:] for every valid (n,m) slot (duplicates accumulate)
__global__ void bead_scatter_kernel(const int* __restrict__ bidc, const float* __restrict__ rec,
                                    float* __restrict__ pos) {
  __shared__ int s_idx[TB * MSLOT];
  const int tid  = threadIdx.x;
  const int blk0 = blockIdx.x * TB;
  const int nb   = (NBEADS_ - blk0 < TB) ? (NBEADS_ - blk0) : TB;
  const int ne   = nb * MSLOT;
  for (int k = tid; k < ne; k += TB)
    asyncCopyB32(bidc + (size_t)blk0 * MSLOT + k, &s_idx[k]);
  asyncWaitZero();
  __syncthreads();
  const int n = blk0 + tid;
  if (n >= NBEADS_) return;
  #pragma unroll
  for (int b = 0; b < BATCH; ++b) {
    const size_t ro = ((size_t)b * NBEADS_ + n) * 3;
    const float rx = rec[ro+0], ry = rec[ro+1], rz = rec[ro+2];
    #pragma unroll
    for (int m = 0; m < MSLOT; ++m) {
      const int idx = s_idx[tid * MSLOT + m];
      if (idx >= 0) {
        float* p = pos + ((size_t)b * NATOM + idx) * 3;
        atomAddF(p + 0, rx);
        atomAddF(p + 1, ry);
        atomAddF(p + 2, rz);
      }
    }
  }
}

__global__ void finalize_kernel(const double* __restrict__ esum, float* __restrict__ outs) {
  if (blockIdx.x == 0 && threadIdx.x == 0) {
    const double eb = 1000.0 * esum[0] / ((double)BATCH * (double)NBOND_);
    const double ea = 150.0  * esum[1] / ((double)BATCH * (double)NANG_);
    const double et = 100.0  * esum[2];
    outs[0] = (float)(eb + ea + et);  // total
    outs[1] = (float)eb;
    outs[2] = (float)ea;
    outs[3] = (float)et;
  }
}

// ---------- launch ----------

extern "C" void kernel_launch(void* const* d_in, const int* in_sizes, int n_in,
                              void* d_out, int out_size, void* d_ws, size_t ws_size,
                              hipStream_t stream) {
  (void)in_sizes; (void)n_in; (void)out_size; (void)ws_size;
  const float* coords     = (const float*)d_in[0];
  const int*   bond_idcs  = (const int*)  d_in[1];
  const float* bond_eq    = (const float*)d_in[2];
  const float* bond_tol   = (const float*)d_in[3];
  const int*   angle_idcs = (const int*)  d_in[4];
  const float* angle_eq   = (const float*)d_in[5];
  const float* angle_tol  = (const float*)d_in[6];
  const int*   omega_idcs = (const int*)  d_in[7];
  const float* omega_eq   = (const float*)d_in[8];
  const float* omega_tol  = (const float*)d_in[9];
  const int*   bb_idcs    = (const int*)  d_in[10];
  const float* bb_w       = (const float*)d_in[11];
  const float* bb_bead    = (const float*)d_in[12];
  const float* dtau       = (const float*)d_in[13];
  // d_in[14] = t (>=1 path only; unused)

  char* ws = (char*)d_ws;
  float*  g    = (float*)ws;                    // 4*500000*3 floats = 24,000,000 B
  float*  rec  = (float*)(ws + 24000000);       // 4*125000*3 floats =  6,000,000 B
  double* esum = (double*)(ws + 30000000);      // 3 doubles (8B aligned)

  float* pos_out  = (float*)d_out;                          // [B,N,3]
  float* scal_out = pos_out + (size_t)BATCH * NATOM * 3;    // total, e_b, e_a, e_t
  float* grad_out = scal_out + 4;                           // [B,N,3]

  zero_kernel<<<(BATCH * NATOM * 3 / 4 + TB - 1) / TB, TB, 0, stream>>>((float4*)g, esum);
  bond_kernel   <<<(NBOND_ + TB - 1) / TB, TB, 0, stream>>>(coords, bond_idcs, bond_eq, bond_tol, g, esum + 0);
  angle_kernel  <<<(NANG_  + TB - 1) / TB, TB, 0, stream>>>(coords, angle_idcs, angle_eq, angle_tol, g, esum + 1);
  torsion_kernel<<<(NTOR_  + TB - 1) / TB, TB, 0, stream>>>(coords, omega_idcs, omega_eq, omega_tol, g, esum + 2);
  update_kernel <<<(BATCH * NATOM + TB - 1) / TB, TB, 0, stream>>>(coords, g, dtau, pos_out, grad_out);
  bead_rec_kernel    <<<(NBEADS_ + TB - 1) / TB, TB, 0, stream>>>(pos_out, bb_idcs, bb_w, bb_bead, rec);
  bead_scatter_kernel<<<(NBEADS_ + TB - 1) / TB, TB, 0, stream>>>(bb_idcs, rec, pos_out);
  finalize_kernel<<<1, 1, 0, stream>>>(esum, scal_out);
}